// WienerDeconvLayer_22308060136148
// MI455X (gfx1250) — compile-verified
//
#include <hip/hip_runtime.h>
#include <math.h>

// ---------------- problem constants ----------------
#define NLOG 520            // padded image size Hp = Wp = 512 + 8
#define NPAD 528            // GEMM-aligned size (33 * 16)
#define NP2  (NPAD*NPAD)
#define BATCH 16
#define ND 4                // wiener filters
#define NF 24               // reg filters
#define RKH 5
#define RKW 5
#define BKH 9
#define BKW 9
#define HH 512
#define WW 512
#define PADT 4

typedef __attribute__((ext_vector_type(2))) float v2f;
typedef __attribute__((ext_vector_type(8))) float v8f;

__device__ __forceinline__ double dphase(long long a) {
  // angle = -2*pi*(a mod N)/N, computed in double for accuracy
  long long m = a % (long long)NLOG;
  return (-2.0 * 3.14159265358979323846 / (double)NLOG) * (double)m;
}

// ---------------- DFT matrices: W (fwd), V = conj(W)/N (inv) ----------------
__global__ void k_dft_mats(float* __restrict__ wr, float* __restrict__ wi,
                           float* __restrict__ vr, float* __restrict__ vi) {
  int t = blockIdx.x * blockDim.x + threadIdx.x;
  if (t >= NP2) return;
  int k = t / NPAD, n = t % NPAD;
  if (k < NLOG && n < NLOG) {
    double ang = dphase((long long)k * (long long)n);
    double c = cos(ang), s = sin(ang);
    wr[t] = (float)c;            wi[t] = (float)s;
    vr[t] = (float)(c / NLOG);   vi[t] = (float)(-s / NLOG);
  } else {
    wr[t] = 0.f; wi[t] = 0.f; vr[t] = 0.f; vi[t] = 0.f;
  }
}

// ---------------- register-blocked WMMA GEMM ----------------
// D = s1 * (A1*B1 + sgn2*A2*B2), all NPAD x NPAD row-major f32.
// Per-wave tile: 32 (M, two 16-tiles) x 48 (N, three 16-tiles) -> 6 v8f accs.
// blockIdx.z = batch; per-operand batch strides. All tile guards are
// wave-uniform so EXEC stays all-ones across every WMMA.
template <bool TWO>
__global__ void k_gemm(const float* __restrict__ A1, const float* __restrict__ B1,
                       const float* __restrict__ A2, const float* __restrict__ B2,
                       float s1, float sgn2,
                       float* __restrict__ D,
                       int sA1, int sB1, int sA2, int sB2, int sD) {
  const int lane = threadIdx.x;           // 0..31
  const int wave = threadIdx.y;           // 0..3
  const int m0 = (blockIdx.y * 4 + wave) * 32;   // two 16-row tiles
  const int n0 = blockIdx.x * 48;                // three 16-col tiles (48 | 528)
  if (m0 >= NPAD) return;                        // wave-uniform exit
  const bool hi = (m0 + 16) < NPAD;              // second M-tile valid? (uniform)
  const int z = blockIdx.z;
  A1 += (size_t)z * sA1;  B1 += (size_t)z * sB1;
  if (TWO) { A2 += (size_t)z * sA2;  B2 += (size_t)z * sB2; }
  D += (size_t)z * sD;

  const int mr   = lane & 15;             // A row / B col / D col within tile
  const int koff = (lane >> 4) << 1;      // lanes 16-31 hold K+2,K+3

  v8f acc[2][3] = {};
  for (int k0 = 0; k0 < NPAD; k0 += 4) {
    // ---- term 1 ----
    v2f a[2], b[3];
    #pragma unroll
    for (int i = 0; i < 2; ++i) {
      if (i == 0 || hi) {
        const float* pa = A1 + (size_t)(m0 + 16 * i + mr) * NPAD + (k0 + koff);
        a[i].x = pa[0]; a[i].y = pa[1];
      }
    }
    #pragma unroll
    for (int j = 0; j < 3; ++j) {
      const float* pb = B1 + (size_t)(k0 + koff) * NPAD + (n0 + 16 * j + mr);
      b[j].x = pb[0]; b[j].y = pb[NPAD];
      __builtin_prefetch(pb + 4 * NPAD, 0, 1);   // stream next B panel rows
    }
    #pragma unroll
    for (int i = 0; i < 2; ++i) {
      if (i == 0 || hi) {
        #pragma unroll
        for (int j = 0; j < 3; ++j)
          acc[i][j] = __builtin_amdgcn_wmma_f32_16x16x4_f32(
              false, a[i], false, b[j], (short)0, acc[i][j], false, false);
      }
    }
    // ---- term 2 (sign folded into B fragment; shares accumulators) ----
    if (TWO) {
      v2f a2[2], b2[3];
      #pragma unroll
      for (int i = 0; i < 2; ++i) {
        if (i == 0 || hi) {
          const float* pa = A2 + (size_t)(m0 + 16 * i + mr) * NPAD + (k0 + koff);
          a2[i].x = pa[0]; a2[i].y = pa[1];
        }
      }
      #pragma unroll
      for (int j = 0; j < 3; ++j) {
        const float* pb = B2 + (size_t)(k0 + koff) * NPAD + (n0 + 16 * j + mr);
        b2[j].x = pb[0] * sgn2; b2[j].y = pb[NPAD] * sgn2;
      }
      #pragma unroll
      for (int i = 0; i < 2; ++i) {
        if (i == 0 || hi) {
          #pragma unroll
          for (int j = 0; j < 3; ++j)
            acc[i][j] = __builtin_amdgcn_wmma_f32_16x16x4_f32(
                false, a2[i], false, b2[j], (short)0, acc[i][j], false, false);
        }
      }
    }
  }
  // C/D layout: lane holds col n = lane&15; VGPR e holds row e + 8*(lane>>4)
  const int rb = (lane >> 4) * 8;
  #pragma unroll
  for (int i = 0; i < 2; ++i) {
    if (i == 0 || hi) {
      #pragma unroll
      for (int j = 0; j < 3; ++j) {
        #pragma unroll
        for (int e = 0; e < 8; ++e) {
          int m = m0 + 16 * i + rb + e;
          D[(size_t)m * NPAD + (n0 + 16 * j + mr)] = s1 * acc[i][j][e];
        }
      }
    }
  }
}

// ---------------- symmetric pad 512 -> 520 (stored in 528 frame, zero pad) ----
__global__ void k_pad_sym(const float* __restrict__ x, float* __restrict__ xp) {
  int t = blockIdx.x * blockDim.x + threadIdx.x;
  if (t >= BATCH * NP2) return;
  int b = t / NP2, r = t % NP2;
  int y = r / NPAD, xx = r % NPAD;
  float v = 0.f;
  if (y < NLOG && xx < NLOG) {
    int sy = y - PADT; if (sy < 0) sy = -1 - sy; else if (sy >= HH) sy = 2 * HH - 1 - sy;
    int sx = xx - PADT; if (sx < 0) sx = -1 - sx; else if (sx >= WW) sx = 2 * WW - 1 - sx;
    v = x[((size_t)b * HH + sy) * WW + sx];
  }
  xp[t] = v;
}

// ---------------- blur OTF + |OTF|^2 (direct 81-tap DFT) ----------------
__global__ void k_otf(const float* __restrict__ bk, float* __restrict__ otr,
                      float* __restrict__ oti, float* __restrict__ b2) {
  int t = blockIdx.x * blockDim.x + threadIdx.x;
  if (t >= NP2) return;
  int k = t / NPAD, l = t % NPAD;
  float re = 0.f, im = 0.f;
  if (k < NLOG && l < NLOG) {
    double sr = 0.0, si = 0.0;
    for (int r = 0; r < BKH; ++r)
      for (int c = 0; c < BKW; ++c) {
        double ang = dphase((long long)k * (r - 4) + (long long)l * (c - 4));
        double w = (double)bk[r * BKW + c];
        sr += w * cos(ang); si += w * sin(ang);
      }
    re = (float)sr; im = (float)si;
  }
  otr[t] = re; oti[t] = im; b2[t] = re * re + im * im;
}

// ---------------- edgetaper profiles (sparse circular autocorrelation) -------
__global__ void k_taper(const float* __restrict__ bk, float* __restrict__ br,
                        float* __restrict__ bc) {
  int t = blockIdx.x * blockDim.x + threadIdx.x;
  if (t >= 2 * NLOG) return;
  int axis = t / NLOG, idx = t % NLOG;
  float p[BKH];
  for (int i = 0; i < BKH; ++i) {
    float s = 0.f;
    for (int j = 0; j < BKW; ++j) s += (axis == 0) ? bk[i * BKW + j] : bk[j * BKW + i];
    p[i] = s;
  }
  const int Lm1 = NLOG - 1;                 // 519
  int d = (idx < Lm1) ? idx : 0;            // entry N-1 wraps to acf[0]
  float acf = 0.f, acf0 = 0.f;
  for (int i = 0; i < BKH; ++i) {
    acf0 += p[i] * p[i];
    int j = (i + d) % Lm1;
    if (j < BKH) acf += p[i] * p[j];
  }
  float v = acf / acf0;                     // acf max is at lag 0
  if (axis == 0) br[idx] = v; else bc[idx] = v;
}

// ---------------- edgetaper blend (in place on xp) ----------------
__global__ void k_blend(float* __restrict__ xp, const float* __restrict__ blur,
                        const float* __restrict__ br, const float* __restrict__ bc) {
  int t = blockIdx.x * blockDim.x + threadIdx.x;
  if (t >= BATCH * NP2) return;
  int r = t % NP2;
  int y = r / NPAD, x = r % NPAD;
  if (y < NLOG && x < NLOG) {
    float tp = (1.f - br[y]) * (1.f - bc[x]);
    xp[t] = tp * xp[t] + (1.f - tp) * blur[t];
  } else {
    xp[t] = 0.f;
  }
}

// ---------------- spectral pointwise: Z = F * otf (or conj(otf)) -------------
__global__ void k_spec_mul(const float* __restrict__ Fr, const float* __restrict__ Fi,
                           const float* __restrict__ orr, const float* __restrict__ oii,
                           float* __restrict__ Zr, float* __restrict__ Zi, int conjO) {
  int t = blockIdx.x * blockDim.x + threadIdx.x;
  if (t >= BATCH * NP2) return;
  int r = t % NP2;
  int y = r / NPAD, x = r % NPAD;
  float zr = 0.f, zi = 0.f;
  if (y < NLOG && x < NLOG) {
    float a = Fr[t], b = Fi[t], c = orr[r], d = oii[r];
    if (conjO) d = -d;
    zr = a * c - b * d;
    zi = a * d + b * c;
  }
  Zr[t] = zr; Zi[t] = zi;
}

// ---------------- weight normalization (zero-mean, L2 over 25, * scale) ------
__global__ void k_wnorm(const float* __restrict__ w, const float* __restrict__ sc,
                        float* __restrict__ wn) {
  int t = blockIdx.x * blockDim.x + threadIdx.x;
  if (t >= ND * NF) return;
  const float* p = w + (size_t)t * RKH * RKW;
  float m = 0.f;
  for (int i = 0; i < RKH * RKW; ++i) m += p[i];
  m *= (1.f / (RKH * RKW));
  float s = 0.f;
  for (int i = 0; i < RKH * RKW; ++i) { float v = p[i] - m; s += v * v; }
  float inv = sc[t] / sqrtf(s);
  for (int i = 0; i < RKH * RKW; ++i) wn[(size_t)t * 25 + i] = (p[i] - m) * inv;
}

// ---------------- denom = |otf|^2 + exp(alpha_d) * sum_f |G_df|^2 ------------
__global__ void k_den(const float* __restrict__ wn, const float* __restrict__ alpha,
                      const float* __restrict__ b2, float* __restrict__ den) {
  int t = blockIdx.x * blockDim.x + threadIdx.x;
  if (t >= ND * NP2) return;
  int d = t / NP2, r = t % NP2;
  int k = r / NPAD, l = r % NPAD;
  if (!(k < NLOG && l < NLOG)) { den[t] = 1.f; return; }
  float cs[25], sn[25];
  for (int i = 0; i < RKH; ++i)
    for (int j = 0; j < RKW; ++j) {
      double ang = dphase((long long)k * (i - 2) + (long long)l * (j - 2));
      cs[i * 5 + j] = (float)cos(ang);
      sn[i * 5 + j] = (float)sin(ang);
    }
  float S = 0.f;
  for (int f = 0; f < NF; ++f) {
    const float* p = wn + (size_t)(d * NF + f) * 25;
    float ar = 0.f, ai = 0.f;
    for (int i = 0; i < 25; ++i) { ar += p[i] * cs[i]; ai += p[i] * sn[i]; }
    S += ar * ar + ai * ai;
  }
  den[t] = b2[r] + __expf(alpha[d]) * S;
}

// ---------------- Wiener division per (b,d): X = Y / den ----------------
__global__ void k_xf(const float* __restrict__ Yr, const float* __restrict__ Yi,
                     const float* __restrict__ den, float* __restrict__ Xr,
                     float* __restrict__ Xi) {
  int t = blockIdx.x * blockDim.x + threadIdx.x;
  if (t >= NP2) return;
  int y = t / NPAD, x = t % NPAD;
  float xr = 0.f, xi = 0.f;
  if (y < NLOG && x < NLOG) {
    float dn = den[t];
    xr = Yr[t] / dn; xi = Yi[t] / dn;
  }
  Xr[t] = xr; Xi[t] = xi;
}

// ---------------- crop 520->512 region out of 528 frame ----------------
__global__ void k_crop(const float* __restrict__ src, float* __restrict__ dst) {
  int t = blockIdx.x * blockDim.x + threadIdx.x;
  if (t >= HH * WW) return;
  int y = t / WW, x = t % WW;
  dst[t] = src[(size_t)(y + PADT) * NPAD + (x + PADT)];
}

// ---------------- cvar reduction + colored-noise std ----------------
__global__ void k_zero(float* __restrict__ p, int n) {
  int t = blockIdx.x * blockDim.x + threadIdx.x;
  if (t < n) p[t] = 0.f;
}

__global__ void k_cvar(const float* __restrict__ b2, const float* __restrict__ den,
                       float* __restrict__ cvar) {
  __shared__ float sm[256];
  int d = blockIdx.y;
  int r = blockIdx.x * 256 + threadIdx.x;
  float v = 0.f;
  if (r < NP2) {
    int y = r / NPAD, x = r % NPAD;
    if (y < NLOG && x < NLOG) {
      float dn = den[(size_t)d * NP2 + r];
      v = b2[r] / (dn * dn);
    }
  }
  sm[threadIdx.x] = v;
  __syncthreads();
  for (int s = 128; s > 0; s >>= 1) {
    if (threadIdx.x < s) sm[threadIdx.x] += sm[threadIdx.x + s];
    __syncthreads();
  }
  if (threadIdx.x == 0) atomicAdd(&cvar[d], sm[0]);
}

__global__ void k_cstdn(const float* __restrict__ stdn, const float* __restrict__ cvar,
                        float* __restrict__ out) {
  int t = blockIdx.x * blockDim.x + threadIdx.x;
  if (t >= BATCH * ND) return;
  int b = t / ND, d = t % ND;
  float cv = cvar[d] * (1.f / ((float)NLOG * (float)NLOG));
  out[t] = sqrtf(stdn[b] * stdn[b] * cv);
}

// =====================================================================
extern "C" void kernel_launch(void* const* d_in, const int* in_sizes, int n_in,
                              void* d_out, int out_size, void* d_ws, size_t ws_size,
                              hipStream_t stream) {
  const float* x     = (const float*)d_in[0];   // (16,1,512,512)
  const float* bk    = (const float*)d_in[1];   // (9,9)
  const float* stdn  = (const float*)d_in[2];   // (16)
  const float* cw    = (const float*)d_in[3];   // (4,24,1,5,5)
  const float* scale = (const float*)d_in[4];   // (4,24)
  const float* alpha = (const float*)d_in[5];   // (4,1)

  float* out      = (float*)d_out;
  float* out_tail = out + (size_t)BATCH * ND * HH * WW;

  // ---- bump allocator over workspace ----
  char* base = (char*)d_ws;
  size_t off = 0;
  auto alloc = [&](size_t nfloats) -> float* {
    float* p = (float*)(base + off);
    off += ((nfloats * sizeof(float) + 255) / 256) * 256;
    return p;
  };
  float* Wfr = alloc(NP2);  float* Wfi = alloc(NP2);
  float* Vr  = alloc(NP2);  float* Vi  = alloc(NP2);
  float* xp  = alloc((size_t)BATCH * NP2);
  float* Tr  = alloc((size_t)BATCH * NP2);  float* Ti  = alloc((size_t)BATCH * NP2);
  float* Fr  = alloc((size_t)BATCH * NP2);  float* Fi  = alloc((size_t)BATCH * NP2);
  float* blr = alloc((size_t)BATCH * NP2);
  float* otr = alloc(NP2);  float* oti = alloc(NP2);  float* b2 = alloc(NP2);
  float* den = alloc((size_t)ND * NP2);
  float* br  = alloc(NPAD); float* bc  = alloc(NPAD);
  float* wn  = alloc((size_t)ND * NF * 25);
  float* cvar = alloc(16);
  float* Xr  = alloc(NP2);  float* Xi = alloc(NP2);
  float* Ur  = alloc(NP2);  float* Ui = alloc(NP2);
  float* ot  = alloc(NP2);
  (void)ws_size; (void)in_sizes; (void)n_in; (void)out_size;

  auto LB = [](size_t n) { return dim3((unsigned)((n + 255) / 256)); };
  const dim3 TB(256);
  const dim3 GB(32, 4);                               // 4 waves of 32 lanes
  // per-wave tile 32x48; block covers 128(M) x 48(N); 528 = 11*48 in N
  auto GG = [](int nb) { return dim3(NPAD / 48, (NPAD + 127) / 128, nb); };

  auto G1 = [&](const float* A, int sA, const float* Bm, int sB, float s,
                float* Dst, int sD, int nb) {
    k_gemm<false><<<GG(nb), GB, 0, stream>>>(A, Bm, nullptr, nullptr, s, 0.f,
                                             Dst, sA, sB, 0, 0, sD);
  };
  auto G2 = [&](const float* A1, int sA1, const float* B1, int sB1, float s1,
                const float* A2, int sA2, const float* B2, int sB2, float s2,
                float* Dst, int sD, int nb) {
    k_gemm<true><<<GG(nb), GB, 0, stream>>>(A1, B1, A2, B2, s1, s2 / s1,
                                            Dst, sA1, sB1, sA2, sB2, sD);
  };

  // ---- setup ----
  k_dft_mats<<<LB(NP2), TB, 0, stream>>>(Wfr, Wfi, Vr, Vi);
  k_pad_sym<<<LB((size_t)BATCH * NP2), TB, 0, stream>>>(x, xp);
  k_otf<<<LB(NP2), TB, 0, stream>>>(bk, otr, oti, b2);
  k_taper<<<LB(2 * NLOG), TB, 0, stream>>>(bk, br, bc);
  k_wnorm<<<LB(ND * NF), TB, 0, stream>>>(cw, scale, wn);
  k_den<<<LB((size_t)ND * NP2), TB, 0, stream>>>(wn, alpha, b2, den);

  // ---- forward FFT of xp: T = xp*W ; F = W*T (complex) ----
  G1(xp, NP2, Wfr, 0, 1.f, Tr, NP2, BATCH);
  G1(xp, NP2, Wfi, 0, 1.f, Ti, NP2, BATCH);
  G2(Wfr, 0, Tr, NP2, 1.f, Wfi, 0, Ti, NP2, -1.f, Fr, NP2, BATCH);
  G2(Wfr, 0, Ti, NP2, 1.f, Wfi, 0, Tr, NP2,  1.f, Fi, NP2, BATCH);

  // ---- blurred = real(ifft2(F * otf)) ----
  k_spec_mul<<<LB((size_t)BATCH * NP2), TB, 0, stream>>>(Fr, Fi, otr, oti, Tr, Ti, 0); // Z in T
  G2(Tr, NP2, Vr, 0, 1.f, Ti, NP2, Vi, 0, -1.f, Fr, NP2, BATCH);   // U in F
  G2(Tr, NP2, Vi, 0, 1.f, Ti, NP2, Vr, 0,  1.f, Fi, NP2, BATCH);
  G2(Vr, 0, Fr, NP2, 1.f, Vi, 0, Fi, NP2, -1.f, blr, NP2, BATCH);  // real part only

  // ---- edgetaper blend (xp <- taper*xp + (1-taper)*blurred) ----
  k_blend<<<LB((size_t)BATCH * NP2), TB, 0, stream>>>(xp, blr, br, bc);

  // ---- forward FFT of blended xp ----
  G1(xp, NP2, Wfr, 0, 1.f, Tr, NP2, BATCH);
  G1(xp, NP2, Wfi, 0, 1.f, Ti, NP2, BATCH);
  G2(Wfr, 0, Tr, NP2, 1.f, Wfi, 0, Ti, NP2, -1.f, Fr, NP2, BATCH);
  G2(Wfr, 0, Ti, NP2, 1.f, Wfi, 0, Tr, NP2,  1.f, Fi, NP2, BATCH);

  // ---- Y = conj(otf) * F   (stored in T) ----
  k_spec_mul<<<LB((size_t)BATCH * NP2), TB, 0, stream>>>(Fr, Fi, otr, oti, Tr, Ti, 1);

  // ---- colored-noise variance ----
  k_zero<<<dim3(1), TB, 0, stream>>>(cvar, 16);
  k_cvar<<<dim3((NP2 + 255) / 256, ND), TB, 0, stream>>>(b2, den, cvar);
  k_cstdn<<<LB(BATCH * ND), TB, 0, stream>>>(stdn, cvar, out_tail);

  // ---- per (b,d): X = Y/den ; out = real(ifft2(X)) cropped ----
  for (int b = 0; b < BATCH; ++b) {
    for (int d = 0; d < ND; ++d) {
      k_xf<<<LB(NP2), TB, 0, stream>>>(Tr + (size_t)b * NP2, Ti + (size_t)b * NP2,
                                       den + (size_t)d * NP2, Xr, Xi);
      G2(Xr, 0, Vr, 0, 1.f, Xi, 0, Vi, 0, -1.f, Ur, 0, 1);
      G2(Xr, 0, Vi, 0, 1.f, Xi, 0, Vr, 0,  1.f, Ui, 0, 1);
      G2(Vr, 0, Ur, 0, 1.f, Vi, 0, Ui, 0, -1.f, ot, 0, 1);        // real part only
      k_crop<<<LB((size_t)HH * WW), TB, 0, stream>>>(
          ot, out + ((size_t)b * ND + d) * HH * WW);
    }
  }
}